// GroupedGemmLayer_22617297781339
// MI455X (gfx1250) — compile-verified
//
#include <hip/hip_runtime.h>

typedef __attribute__((ext_vector_type(2))) float v2f;
typedef __attribute__((ext_vector_type(8))) float v8f;
typedef __attribute__((ext_vector_type(4))) int   v4i;

#define K_DIM 1024
#define N_DIM 512
#define BM 128
#define BN 128
#define KC 16
#define NCHUNK (K_DIM / KC)   // 64
#define MAX_TILES 24          // ceil(3072/128)
#define GROUPS 64
#define ASTR 20               // A LDS row stride (floats): spacing-4 bank spread -> conflict-free b64 frags
#define BSTRP 288             // B LDS row-pair stride (floats): +32-bank shift between lane halves

#if __has_builtin(__builtin_amdgcn_global_load_async_to_lds_b128)
#define HAVE_ASYNC_LDS 1
#else
#define HAVE_ASYNC_LDS 0
#endif

__device__ __forceinline__ void wait_asynccnt0() {
#if __has_builtin(__builtin_amdgcn_s_wait_asynccnt)
  __builtin_amdgcn_s_wait_asynccnt(0);
#else
  asm volatile("s_wait_asynccnt 0x0" ::: "memory");
#endif
}

__global__ __launch_bounds__(256, 1) void grouped_gemm_wmma_f32(
    const float* __restrict__ A,      // [T, K]
    const float* __restrict__ B,      // [G, K, N]
    const int*   __restrict__ sizes,  // [G]
    const int*   __restrict__ offs,   // [G]
    float*       __restrict__ out)    // [T, N]
{
  __shared__ float As[2][BM * ASTR];          // [row][k]
  __shared__ float Bs[2][(KC / 2) * BSTRP];   // [kpair][2n + (k&1)] interleaved

  const int g    = blockIdx.x / MAX_TILES;
  const int mt   = blockIdx.x % MAX_TILES;
  const int size = sizes[g];
  if (mt * BM >= size) return;
  const int off  = offs[g];
  const int row0 = mt * BM;
  const int n0   = blockIdx.y * BN;

  const int t    = threadIdx.x;
  const int lane = t & 31;
  const int wid  = t >> 5;
  const int wm   = wid & 3;                   // rows wm*32 (2 M-tiles)
  const int wn   = wid >> 2;                  // cols wn*64 (4 N-tiles)
  const int half = lane >> 4;
  const int mn   = lane & 15;

  const float* __restrict__ Bg = B + (size_t)g * ((size_t)K_DIM * N_DIM) + n0;

  // ---- loop-invariant staging addresses ----
  // A tile 128x16: 2 rows x float4 per thread
  const int a_tr = t >> 2;                    // 0..63
  const int a_tc = (t & 3) * 4;               // 0..12
  int gr0 = row0 + a_tr;
  int gr1 = row0 + 64 + a_tr;
  const float* aSrc0 = A + (size_t)(off + (gr0 < size ? gr0 : size - 1)) * K_DIM + a_tc;
  const float* aSrc1 = A + (size_t)(off + (gr1 < size ? gr1 : size - 1)) * K_DIM + a_tc;
  // B tile 16x128: one row-pair x float4 per thread
  const int b_rp = t >> 5;                    // 0..7
  const int b_nc = (t & 31) * 4;              // 0..124
  const float* bSrc0 = Bg + (size_t)(2 * b_rp) * N_DIM + b_nc;
  const float* bSrc1 = bSrc0 + N_DIM;
  // LDS staging destinations (element offsets within a buffer)
  const int aDst0 = a_tr * ASTR + a_tc;
  const int aDst1 = (64 + a_tr) * ASTR + a_tc;
  const int bDst  = b_rp * BSTRP + 2 * b_nc;
  // loop-invariant fragment read offsets
  const int aOff = (wm * 32 + mn) * ASTR + 2 * half;      // + mi*16*ASTR + k
  const int bOff = half * BSTRP + 2 * (wn * 64 + mn);     // + ks*2*BSTRP + ni*32

  float4 rb0, rb1;
#if !HAVE_ASYNC_LDS
  float4 ra0, ra1;
#endif

#if HAVE_ASYNC_LDS
  auto stageA = [&](const float* s0, const float* s1, int p) {
    __builtin_amdgcn_global_load_async_to_lds_b128(
        (v4i*)s0, (v4i*)&As[p][aDst0], 0, 0);
    __builtin_amdgcn_global_load_async_to_lds_b128(
        (v4i*)s1, (v4i*)&As[p][aDst1], 0, 0);
  };
#endif

  v8f acc[2][4];
#pragma unroll
  for (int mi = 0; mi < 2; ++mi)
#pragma unroll
    for (int ni = 0; ni < 4; ++ni)
      acc[mi][ni] = (v8f)(0.0f);

  // ---- prologue: chunk 0 in flight ----
#if HAVE_ASYNC_LDS
  stageA(aSrc0, aSrc1, 0);
#else
  ra0 = *(const float4*)aSrc0;
  ra1 = *(const float4*)aSrc1;
#endif
  rb0 = *(const float4*)bSrc0;
  rb1 = *(const float4*)bSrc1;
  aSrc0 += KC; aSrc1 += KC; bSrc0 += KC * N_DIM; bSrc1 += KC * N_DIM;

  for (int c = 0; c < NCHUNK; ++c) {
    const int p = c & 1;
    // commit B chunk c (interleaved) into buffer p
    {
      float* dst = &Bs[p][bDst];
      float4 lo = {rb0.x, rb1.x, rb0.y, rb1.y};
      float4 hi = {rb0.z, rb1.z, rb0.w, rb1.w};
      *(float4*)(dst)     = lo;
      *(float4*)(dst + 4) = hi;
    }
#if HAVE_ASYNC_LDS
    wait_asynccnt0();                       // A chunk c landed in LDS
#else
    *(float4*)&As[p][aDst0] = ra0;
    *(float4*)&As[p][aDst1] = ra1;
#endif
    __syncthreads();                        // publish buffer p

    if (c + 1 < NCHUNK) {                   // prefetch chunk c+1 -> buffer p^1
#if HAVE_ASYNC_LDS
      stageA(aSrc0, aSrc1, p ^ 1);
#else
      ra0 = *(const float4*)aSrc0;
      ra1 = *(const float4*)aSrc1;
#endif
      rb0 = *(const float4*)bSrc0;
      rb1 = *(const float4*)bSrc1;
      aSrc0 += KC; aSrc1 += KC; bSrc0 += KC * N_DIM; bSrc1 += KC * N_DIM;
    }

#pragma unroll
    for (int ks = 0; ks < KC / 4; ++ks) {
      const int k = ks * 4;
      v2f afr[2];
      v2f bfr[4];
#pragma unroll
      for (int mi = 0; mi < 2; ++mi)
        afr[mi] = *(const v2f*)&As[p][aOff + mi * 16 * ASTR + k];
#pragma unroll
      for (int ni = 0; ni < 4; ++ni)
        bfr[ni] = *(const v2f*)&Bs[p][bOff + ks * 2 * BSTRP + ni * 32];
#pragma unroll
      for (int mi = 0; mi < 2; ++mi)
#pragma unroll
        for (int ni = 0; ni < 4; ++ni)
          acc[mi][ni] = __builtin_amdgcn_wmma_f32_16x16x4_f32(
              false, afr[mi], false, bfr[ni],
              (short)0, acc[mi][ni], false, false);
    }
  }

  // C/D layout: lane -> col = mn, VGPR v -> row = 8*half + v
#pragma unroll
  for (int mi = 0; mi < 2; ++mi) {
#pragma unroll
    for (int v = 0; v < 8; ++v) {
      int row = row0 + wm * 32 + mi * 16 + half * 8 + v;
      if (row < size) {
        float* dst = out + (size_t)(off + row) * N_DIM + n0 + wn * 64;
#pragma unroll
        for (int ni = 0; ni < 4; ++ni)
          dst[ni * 16 + mn] = acc[mi][ni][v];
      }
    }
  }
}

extern "C" void kernel_launch(void* const* d_in, const int* in_sizes, int n_in,
                              void* d_out, int out_size, void* d_ws, size_t ws_size,
                              hipStream_t stream) {
  const float* a     = (const float*)d_in[0];
  const float* b     = (const float*)d_in[1];
  const int*   sizes = (const int*)d_in[2];
  const int*   offs  = (const int*)d_in[3];
  float* out = (float*)d_out;

  dim3 grid(GROUPS * MAX_TILES, N_DIM / BN, 1);
  dim3 block(256, 1, 1);
  grouped_gemm_wmma_f32<<<grid, block, 0, stream>>>(a, b, sizes, offs, out);
}